// SpectralResidualAugmentation_16423954940091
// MI455X (gfx1250) — compile-verified
//
#include <hip/hip_runtime.h>
#include <math.h>

// ---------------------------------------------------------------------------
// Spectral residual saliency augmentation, MI455X / gfx1250.
//   img [16,2048,2048] f32 -> out = img * minmax(|ifft2(scale*fft2(img))|)
// 2D FFT = row FFT + transpose + row FFT. Each 2048-pt FFT: radix-16 first
// stage done as complex 16x16 @ 16x128 matmul on V_WMMA_F32_16X16X4_F32,
// twiddle, then 16 independent 128-pt radix-2 Stockham FFTs in LDS.
// One wave32 owns one row; 2 rows / 64-thread block; 66 KB dynamic LDS.
//
// Workspace layout (requires ws_size >= 1610612864 bytes):
//   [0        , 512MB) complex buffer A
//   [512MB    ,   1GB) complex buffer B
//   [1GB      ,1.25GB) float buffer L
//   [1.25GB   ,1.50GB) float buffer T
//   [1.50GB   , +128B) per-channel min/max (uint bits)
// ---------------------------------------------------------------------------

typedef float v2f __attribute__((ext_vector_type(2)));
typedef float v8f __attribute__((ext_vector_type(8)));

static __device__ __forceinline__ v8f wmma_f32_16x16x4(v2f a, v2f b, v8f c) {
  // D = A(16x4) * B(4x16) + C(16x16), all f32.  8-arg form like the f16 op:
  // (neg_a, A, neg_b, B, c_mod, C, reuse_a, reuse_b)
  return __builtin_amdgcn_wmma_f32_16x16x4_f32(false, a, false, b, (short)0, c,
                                               false, false);
}

#define FFT_N 2048
#define ROWS_PER_BLOCK 2
#define LDS_ROW_F (4 * 2064)  // 4 arrays (reS,imS,reD,imD), 2048+16 pad each

static __device__ __forceinline__ int pidx(int i) { return i + (i >> 7); }

// IN_REAL: input is float (imag=0).  OUT_ABS: output |z| as float.
// INV: conjugate twiddles + 1/2048 scale (one pass of ifft normalization).
template <bool IN_REAL, bool OUT_ABS, bool INV>
__global__ __launch_bounds__(64) void fft_rows_kernel(
    const void* __restrict__ inp, void* __restrict__ outp) {
  extern __shared__ float smem[];
  const int wave = threadIdx.x >> 5;
  const int lane = threadIdx.x & 31;
  const size_t row = (size_t)blockIdx.x * ROWS_PER_BLOCK + wave;

  float* base = smem + wave * LDS_ROW_F;
  float* reS = base;
  float* imS = base + 2064;
  float* reD = base + 2 * 2064;
  float* imD = base + 3 * 2064;

  const float sgn = INV ? 1.0f : -1.0f;
  const float TWO_PI = 6.283185307179586f;

  // ---- load row (natural order: M[n1][n2] = x[n1*128+n2] = x[n]) ----------
  if (IN_REAL) {
    const float* in = (const float*)inp + row * FFT_N;
    for (int m = 0; m < 64; ++m) {
      int n = lane + 32 * m;
      reS[pidx(n)] = in[n];
      imS[pidx(n)] = 0.0f;
    }
  } else {
    const float2* in = (const float2*)inp + row * FFT_N;
    for (int m = 0; m < 64; ++m) {
      int n = lane + 32 * m;
      float2 v = in[n];
      reS[pidx(n)] = v.x;
      imS[pidx(n)] = v.y;
    }
  }
  __syncthreads();

  // ---- build DFT16 A-matrix chunks in registers ---------------------------
  // A[m][k] = exp(sgn*i*2pi*m*k/16); ISA A layout: lane m=lane&15,
  // VGPR0 <-> K = 2*(lane>>4), VGPR1 <-> K+1 (per 16x4 chunk c => +4c).
  const int mA = lane & 15;
  const int half = lane >> 4;
  v2f ar[4], ai[4], ani[4];
#pragma unroll
  for (int c = 0; c < 4; ++c) {
    int k0 = 4 * c + 2 * half;
    float a0 = sgn * TWO_PI * (float)(mA * k0) * (1.0f / 16.0f);
    float a1 = sgn * TWO_PI * (float)(mA * (k0 + 1)) * (1.0f / 16.0f);
    float s0, c0, s1, c1;
    __sincosf(a0, &s0, &c0);
    __sincosf(a1, &s1, &c1);
    ar[c] = v2f{c0, c1};
    ai[c] = v2f{s0, s1};
    ani[c] = v2f{-s0, -s1};
  }

  // ---- stage 1: Y = DFT16 x M  (8 column tiles of 16), + twiddle ----------
  const int ncol = lane & 15;
  for (int t = 0; t < 8; ++t) {
    int n2 = t * 16 + ncol;
    v2f br[4], bi[4];
#pragma unroll
    for (int c = 0; c < 4; ++c) {
      int k0 = 4 * c + 2 * half;  // B VGPR0 <-> K=2*half, VGPR1 <-> K+1
      br[c] = v2f{reS[pidx(k0 * 128 + n2)], reS[pidx((k0 + 1) * 128 + n2)]};
      bi[c] = v2f{imS[pidx(k0 * 128 + n2)], imS[pidx((k0 + 1) * 128 + n2)]};
    }
    v8f accr = {0.f, 0.f, 0.f, 0.f, 0.f, 0.f, 0.f, 0.f};
    v8f acci = {0.f, 0.f, 0.f, 0.f, 0.f, 0.f, 0.f, 0.f};
#pragma unroll
    for (int c = 0; c < 4; ++c) accr = wmma_f32_16x16x4(ar[c], br[c], accr);
#pragma unroll
    for (int c = 0; c < 4; ++c) accr = wmma_f32_16x16x4(ani[c], bi[c], accr);
#pragma unroll
    for (int c = 0; c < 4; ++c) acci = wmma_f32_16x16x4(ar[c], bi[c], acci);
#pragma unroll
    for (int c = 0; c < 4; ++c) acci = wmma_f32_16x16x4(ai[c], br[c], acci);

    // twiddle exp(sgn*i*2pi*k1*n2/2048), scatter to Y[k1*128+n2]
#pragma unroll
    for (int j = 0; j < 8; ++j) {
      int mrow = j + 8 * half;  // D layout: vgpr j, lanes>=16 => M=j+8
      float ang = sgn * TWO_PI * (float)(mrow * n2) * (1.0f / 2048.0f);
      float s, cc;
      __sincosf(ang, &s, &cc);
      float yr = accr[j] * cc - acci[j] * s;
      float yi = accr[j] * s + acci[j] * cc;
      reD[pidx(mrow * 128 + n2)] = yr;
      imD[pidx(mrow * 128 + n2)] = yi;
    }
  }
  __syncthreads();

  // ---- stage 2: 16 independent 128-pt radix-2 Stockham FFTs ---------------
  float *sr = reD, *si = imD, *dr = reS, *di = imS;
  for (int st = 0; st < 7; ++st) {
    int s_t = 1 << st;
    int n_t = 128 >> st;
    for (int it = 0; it < 32; ++it) {
      int g = lane + 32 * it;  // 1024 butterflies per row per stage
      int chunk = g >> 6;
      int bb = g & 63;
      int p_ = bb >> st;
      int q = bb & (s_t - 1);
      int basei = chunk * 128;
      int i0 = basei + bb;        // = base + q + s_t*p_
      int i1 = i0 + 64;           // s_t * (n_t/2) == 64 always
      int j0 = basei + q + 2 * s_t * p_;
      int j1 = j0 + s_t;
      float arr = sr[pidx(i0)], ari = si[pidx(i0)];
      float brr = sr[pidx(i1)], bri = si[pidx(i1)];
      float ang = sgn * TWO_PI * (float)p_ / (float)n_t;
      float ws, wc;
      __sincosf(ang, &ws, &wc);
      float tr = arr - brr, ti = ari - bri;
      dr[pidx(j0)] = arr + brr;
      di[pidx(j0)] = ari + bri;
      dr[pidx(j1)] = tr * wc - ti * ws;
      di[pidx(j1)] = tr * ws + ti * wc;
    }
    __syncthreads();
    float* t0 = sr; sr = dr; dr = t0;
    float* t1 = si; si = di; di = t1;
  }
  // result now in sr/si; X[k1 + 16*k2] = buf[k1*128 + k2]

  const float scale = INV ? (1.0f / 2048.0f) : 1.0f;
  if (OUT_ABS) {
    float* out = (float*)outp + row * FFT_N;
    for (int m = 0; m < 64; ++m) {
      int g = lane + 32 * m;
      int b = (g & 15) * 128 + (g >> 4);
      float vr = sr[pidx(b)] * scale;
      float vi = si[pidx(b)] * scale;
      out[g] = sqrtf(vr * vr + vi * vi);
    }
  } else {
    float2* out = (float2*)outp + row * FFT_N;
    for (int m = 0; m < 64; ++m) {
      int g = lane + 32 * m;
      int b = (g & 15) * 128 + (g >> 4);
      out[g] = make_float2(sr[pidx(b)] * scale, si[pidx(b)] * scale);
    }
  }
}

// ---- complex transpose [C,H,W] -> [C,W,H], 32x32 LDS tiles ----------------
__global__ void transpose_cmplx(const float2* __restrict__ in,
                                float2* __restrict__ out, int H, int W) {
  __shared__ float2 tile[32][33];
  const int c = blockIdx.z;
  const int x0 = blockIdx.x * 32;
  const int y0 = blockIdx.y * 32;
  const size_t plane = (size_t)H * W;
  for (int i = threadIdx.y; i < 32; i += 8) {
    int y = y0 + i, x = x0 + threadIdx.x;
    tile[i][threadIdx.x] = in[(size_t)c * plane + (size_t)y * W + x];
  }
  __syncthreads();
  for (int i = threadIdx.y; i < 32; i += 8) {
    int x = x0 + i, y = y0 + threadIdx.x;
    out[(size_t)c * plane + (size_t)x * H + y] = tile[threadIdx.x][i];
  }
}

// ---- shifted log magnitude: L[c,xs,ys] = log1p(|F[c,(xs+1024)%N,(ys+1024)%N]|)
__global__ void logmag_shift(const float2* __restrict__ F,
                             float* __restrict__ L) {
  int idx = blockIdx.x * blockDim.x + threadIdx.x;
  int c = idx >> 22;
  int rem = idx & ((1 << 22) - 1);
  int xs = rem >> 11, ys = rem & 2047;
  int x = (xs + 1024) & 2047, y = (ys + 1024) & 2047;
  float2 f = F[((size_t)c << 22) + ((size_t)x << 11) + y];
  L[idx] = log1pf(sqrtf(f.x * f.x + f.y * f.y));
}

static __device__ __forceinline__ int refl(int i) {
  i = (i < 0) ? -i : i;            // BORDER_REFLECT_101
  return (i > 2047) ? (4094 - i) : i;
}

__global__ void blur_last(const float* __restrict__ in,
                          float* __restrict__ out) {
  int idx = blockIdx.x * blockDim.x + threadIdx.x;
  int r = idx >> 11;   // combined (c,xs) row index
  int ys = idx & 2047;
  float s = 0.f;
#pragma unroll
  for (int d = -4; d <= 4; ++d) s += in[((size_t)r << 11) + refl(ys + d)];
  out[idx] = s * (1.0f / 9.0f);
}

__global__ void blur_mid(const float* __restrict__ in,
                         float* __restrict__ out) {
  int idx = blockIdx.x * blockDim.x + threadIdx.x;
  int c = idx >> 22;
  int rem = idx & ((1 << 22) - 1);
  int xs = rem >> 11, ys = rem & 2047;
  float s = 0.f;
#pragma unroll
  for (int d = -4; d <= 4; ++d)
    s += in[((size_t)c << 22) + ((size_t)refl(xs + d) << 11) + ys];
  out[idx] = s * (1.0f / 9.0f);
}

// ---- G = expm1(log1p(|F|) - blur_shifted) / |F| * F -----------------------
__global__ void spectral_scale(const float2* __restrict__ F,
                               const float* __restrict__ Bs,
                               float2* __restrict__ G) {
  int idx = blockIdx.x * blockDim.x + threadIdx.x;
  int c = idx >> 22;
  int rem = idx & ((1 << 22) - 1);
  int x = rem >> 11, y = rem & 2047;
  int xs = (x + 1024) & 2047, ys = (y + 1024) & 2047;
  float2 f = F[idx];
  float m = sqrtf(f.x * f.x + f.y * f.y);
  float blur = Bs[((size_t)c << 22) + ((size_t)xs << 11) + ys];
  float s = expm1f(log1pf(m) - blur) / m;
  G[idx] = make_float2(f.x * s, f.y * s);
}

// ---- per-channel min/max (sal >= 0 so uint-bit atomics are order-correct) -
__global__ void mm_init(unsigned* mm) {
  int i = threadIdx.x;
  if (i < 32) mm[i] = (i & 1) ? 0u : 0x7F800000u;  // max:=0, min:=+inf
}

__global__ void mm_reduce(const float* __restrict__ sal, unsigned* mm) {
  __shared__ float smn[256], smx[256];
  const int c = blockIdx.y;
  const int per = (1 << 22) / gridDim.x;
  const size_t base = ((size_t)c << 22) + (size_t)blockIdx.x * per;
  float mn = INFINITY, mx = 0.0f;
  for (int i = threadIdx.x; i < per; i += blockDim.x) {
    float v = sal[base + i];
    mn = fminf(mn, v);
    mx = fmaxf(mx, v);
  }
  smn[threadIdx.x] = mn;
  smx[threadIdx.x] = mx;
  __syncthreads();
  for (int s = 128; s > 0; s >>= 1) {
    if (threadIdx.x < s) {
      smn[threadIdx.x] = fminf(smn[threadIdx.x], smn[threadIdx.x + s]);
      smx[threadIdx.x] = fmaxf(smx[threadIdx.x], smx[threadIdx.x + s]);
    }
    __syncthreads();
  }
  if (threadIdx.x == 0) {
    atomicMin(&mm[2 * c], __float_as_uint(smn[0]));
    atomicMax(&mm[2 * c + 1], __float_as_uint(smx[0]));
  }
}

__global__ void finalize(const float* __restrict__ img,
                         const float* __restrict__ sal,
                         const unsigned* __restrict__ mm,
                         float* __restrict__ out) {
  int idx = blockIdx.x * blockDim.x + threadIdx.x;
  int c = idx >> 22;
  float mn = __uint_as_float(mm[2 * c]);
  float mx = __uint_as_float(mm[2 * c + 1]);
  out[idx] = img[idx] * ((sal[idx] - mn) / (mx - mn));
}

// ---------------------------------------------------------------------------
extern "C" void kernel_launch(void* const* d_in, const int* in_sizes, int n_in,
                              void* d_out, int out_size, void* d_ws,
                              size_t ws_size, hipStream_t stream) {
  (void)in_sizes; (void)n_in; (void)out_size; (void)ws_size;
  const float* img = (const float*)d_in[0];  // [16,2048,2048] f32 (k==9 fixed)
  float* out = (float*)d_out;

  char* ws = (char*)d_ws;
  float2* A = (float2*)(ws);
  float2* B = (float2*)(ws + 536870912ull);
  float* L = (float*)(ws + 1073741824ull);
  float* T = (float*)(ws + 1342177280ull);
  unsigned* MM = (unsigned*)(ws + 1610612736ull);

  const int NROWS = 16 * 2048;            // rows per 1-D FFT pass
  const dim3 fftGrid(NROWS / ROWS_PER_BLOCK), fftBlk(64);
  const size_t smem = (size_t)ROWS_PER_BLOCK * LDS_ROW_F * sizeof(float);
  const int NEL = 16 * 2048 * 2048;       // 67,108,864
  const dim3 eGrid(NEL / 256), eBlk(256);
  const dim3 tGrid(64, 64, 16), tBlk(32, 8);

  // forward 2D FFT: rows, transpose, rows          A = F^T  [C,W,H]
  fft_rows_kernel<true, false, false><<<fftGrid, fftBlk, smem, stream>>>(img, A);
  transpose_cmplx<<<tGrid, tBlk, 0, stream>>>(A, B, 2048, 2048);
  fft_rows_kernel<false, false, false><<<fftGrid, fftBlk, smem, stream>>>(B, A);

  // shifted log-magnitude + separable 9x9 reflect blur
  logmag_shift<<<eGrid, eBlk, 0, stream>>>(A, L);
  blur_last<<<eGrid, eBlk, 0, stream>>>(L, T);
  blur_mid<<<eGrid, eBlk, 0, stream>>>(T, L);

  // spectral residual rescale                      B = G^T
  spectral_scale<<<eGrid, eBlk, 0, stream>>>(A, L, B);

  // inverse 2D FFT: rows, transpose, rows(+abs)    L = sal [C,H,W]
  fft_rows_kernel<false, false, true><<<fftGrid, fftBlk, smem, stream>>>(B, A);
  transpose_cmplx<<<tGrid, tBlk, 0, stream>>>(A, B, 2048, 2048);
  fft_rows_kernel<false, true, true><<<fftGrid, fftBlk, smem, stream>>>(B, L);

  // per-channel min/max normalize, multiply into output
  mm_init<<<1, 32, 0, stream>>>(MM);
  mm_reduce<<<dim3(256, 16), 256, 0, stream>>>(L, MM);
  finalize<<<eGrid, eBlk, 0, stream>>>(img, L, MM, out);
}